// GNNML1_64991445123417
// MI455X (gfx1250) — compile-verified
//
#include <hip/hip_runtime.h>
#include <math.h>

typedef __attribute__((ext_vector_type(2))) float v2f;
typedef __attribute__((ext_vector_type(8))) float v8f;

// ---------------------------------------------------------------------------
// Weight packing: Wcat[k][n], n in [0,160): [0,64)=branch-A, [64,128)=conv,
// [128,144)=mul1, [144,160)=mul2.  All sources are row-major [K, out].
// ---------------------------------------------------------------------------
__global__ void pack_w_kernel(const float* __restrict__ WA,
                              const float* __restrict__ WC,
                              const float* __restrict__ WM1,
                              const float* __restrict__ WM2,
                              float* __restrict__ dst, int K) {
    int i = blockIdx.x * blockDim.x + threadIdx.x;
    int total = K * 160;
    if (i >= total) return;
    int k = i / 160, n = i % 160;
    float v;
    if (n < 64)       v = WA [k * 64 + n];
    else if (n < 128) v = WC [k * 64 + (n - 64)];
    else if (n < 144) v = WM1[k * 16 + (n - 128)];
    else              v = WM2[k * 16 + (n - 144)];
    dst[i] = v;
}

// ---------------------------------------------------------------------------
// WMMA f32 GEMM:  Out[N,160] = A[N,K] @ W[K,160]   (K % 4 == 0, N % 16 == 0)
// One wave per 16-row tile; 10 col-tiles of 16; K in steps of 4 via
// V_WMMA_F32_16X16X4_F32 (full f32 precision).
// A-frag (16x4):  lane l -> row l&15,  a[v] = A[row][k0 + 2*(l>>4) + v]
// B-frag (4x16):  lane l -> col l&15,  b[v] = W[k0 + 2*(l>>4) + v][col]
// D (16x16 f32):  lane l, vgpr v -> M = v + 8*(l>>4), N = l&15
// ---------------------------------------------------------------------------
__global__ void gemm_wmma_kernel(const float* __restrict__ A, int K,
                                 const float* __restrict__ W,
                                 float* __restrict__ Out, int nTiles) {
    int wave = threadIdx.x >> 5;
    int lane = threadIdx.x & 31;
    int tile = blockIdx.x * (blockDim.x >> 5) + wave;
    if (tile >= nTiles) return;                 // wave-uniform: EXEC stays all-1s

    int m    = lane & 15;
    int half = lane >> 4;

    const float* arow = A + (size_t)(tile * 16 + m) * (size_t)K;

    v8f acc[10];
    const v8f zero = {0.f, 0.f, 0.f, 0.f, 0.f, 0.f, 0.f, 0.f};
#pragma unroll
    for (int ct = 0; ct < 10; ++ct) acc[ct] = zero;

    for (int k0 = 0; k0 < K; k0 += 4) {
        int ka = k0 + 2 * half;
        v2f a;
        a.x = arow[ka];
        a.y = arow[ka + 1];
        const float* wr = W + (size_t)ka * 160;
#pragma unroll
        for (int ct = 0; ct < 10; ++ct) {
            v2f b;
            b.x = wr[ct * 16 + m];
            b.y = wr[160 + ct * 16 + m];
            acc[ct] = __builtin_amdgcn_wmma_f32_16x16x4_f32(
                false, a, false, b, (short)0, acc[ct], false, false);
        }
    }

    float* obase = Out + (size_t)tile * 16 * 160;
#pragma unroll
    for (int ct = 0; ct < 10; ++ct) {
#pragma unroll
        for (int v = 0; v < 8; ++v) {
            obase[(size_t)(v + 8 * half) * 160 + ct * 16 + m] = acc[ct][v];
        }
    }
}

// ---------------------------------------------------------------------------
// Build feature rows from raw GEMM output:
//   x[0:64)    = relu(h[0:64) + bA)
//   x[64:128)  = 0                      (scatter target, finished later)
//   x[128:144) = relu(h[128+c]+bM1)*relu(h[144+c]+bM2)
// ---------------------------------------------------------------------------
__global__ void build_x_kernel(const float* __restrict__ H,
                               const float* __restrict__ bA,
                               const float* __restrict__ bM1,
                               const float* __restrict__ bM2,
                               float* __restrict__ X, int N) {
    int n = blockIdx.x * blockDim.x + threadIdx.x;
    if (n >= N) return;
    const float* h = H + (size_t)n * 160;
    float*       x = X + (size_t)n * 144;
#pragma unroll 4
    for (int c = 0; c < 64; ++c) x[c] = fmaxf(h[c] + bA[c], 0.f);
#pragma unroll 4
    for (int c = 0; c < 64; ++c) x[64 + c] = 0.f;
#pragma unroll 4
    for (int c = 0; c < 16; ++c) {
        float u = fmaxf(h[128 + c] + bM1[c], 0.f);
        float v = fmaxf(h[144 + c] + bM2[c], 0.f);
        x[128 + c] = u * v;
    }
}

// ---------------------------------------------------------------------------
// Edge scatter: for each edge (s->d), X[d][64+c] += H[s][64+c], c in [0,64).
// 16 lanes per edge, float4 gather (16B aligned), 4 f32 atomic adds.
// ---------------------------------------------------------------------------
__global__ void scatter_add_kernel(const float* __restrict__ H,
                                   const int* __restrict__ src,
                                   const int* __restrict__ dst,
                                   float* __restrict__ X, int E) {
    int t = blockIdx.x * blockDim.x + threadIdx.x;
    int e = t >> 4;
    if (e >= E) return;
    int c4 = (t & 15) * 4;
    int s = src[e];
    int d = dst[e];
    const float4 v = *(const float4*)(H + (size_t)s * 160 + 64 + c4);
    float* out = X + (size_t)d * 144 + 64 + c4;
    atomicAdd(out + 0, v.x);
    atomicAdd(out + 1, v.y);
    atomicAdd(out + 2, v.z);
    atomicAdd(out + 3, v.w);
}

// x[64+c] = relu(x[64+c] + bConv[c])
__global__ void finish_conv_kernel(float* __restrict__ X,
                                   const float* __restrict__ b, int N) {
    int t = blockIdx.x * blockDim.x + threadIdx.x;
    if (t >= N * 64) return;
    int n = t >> 6, c = t & 63;
    float* p = X + (size_t)n * 144 + 64 + c;
    *p = fmaxf(*p + b[c], 0.f);
}

// ---------------------------------------------------------------------------
// Pooling
// ---------------------------------------------------------------------------
__device__ inline void atomicMaxFloat(float* addr, float value) {
    if (value >= 0.f)
        atomicMax((int*)addr, __float_as_int(value));
    else
        atomicMin((unsigned int*)addr, __float_as_uint(value));
}

__global__ void pool_init_kernel(float* __restrict__ sums,
                                 float* __restrict__ maxs,
                                 int* __restrict__ cnt) {
    int t = blockIdx.x * blockDim.x + threadIdx.x;
    if (t < 256 * 144) {
        sums[t] = 0.f;
        maxs[t] = -__builtin_inff();
    }
    if (t < 256) cnt[t] = 0;
}

__global__ void pool_accum_kernel(const float* __restrict__ X,
                                  const int* __restrict__ batch,
                                  float* __restrict__ sums,
                                  float* __restrict__ maxs,
                                  int* __restrict__ cnt, int N) {
    int t = blockIdx.x * blockDim.x + threadIdx.x;
    if (t >= N * 144) return;
    int n = t / 144, c = t - n * 144;
    int g = batch[n];
    float v = X[(size_t)n * 144 + c];
    atomicAdd(&sums[g * 144 + c], v);
    atomicMaxFloat(&maxs[g * 144 + c], v);
    if (c == 0) atomicAdd(&cnt[g], 1);
}

// ---------------------------------------------------------------------------
// Head: g = [mean(144) | max(144)],  out = log_softmax(g @ W[288,2] + b)
// ---------------------------------------------------------------------------
__global__ void head_kernel(const float* __restrict__ sums,
                            const float* __restrict__ maxs,
                            const int* __restrict__ cnt,
                            const float* __restrict__ W,
                            const float* __restrict__ b,
                            float* __restrict__ out, int G) {
    int g = blockIdx.x * blockDim.x + threadIdx.x;
    if (g >= G) return;
    float c = fmaxf((float)cnt[g], 1.f);
    float inv = 1.f / c;
    float z0 = b[0], z1 = b[1];
    for (int k = 0; k < 144; ++k) {
        float mv = sums[g * 144 + k] * inv;
        z0 += mv * W[k * 2 + 0];
        z1 += mv * W[k * 2 + 1];
    }
    for (int k = 0; k < 144; ++k) {
        float xv = maxs[g * 144 + k];
        z0 += xv * W[(144 + k) * 2 + 0];
        z1 += xv * W[(144 + k) * 2 + 1];
    }
    float mx  = fmaxf(z0, z1);
    float lse = mx + logf(expf(z0 - mx) + expf(z1 - mx));
    out[g * 2 + 0] = z0 - lse;
    out[g * 2 + 1] = z1 - lse;
}

// ---------------------------------------------------------------------------
static inline size_t align256(size_t x) { return (x + 255) & ~(size_t)255; }

extern "C" void kernel_launch(void* const* d_in, const int* in_sizes, int n_in,
                              void* d_out, int out_size, void* d_ws, size_t ws_size,
                              hipStream_t stream) {
    const int N = in_sizes[0] / 4;     // 100000 nodes (4 input feats)
    const int E = in_sizes[1] / 2;     // 1.6M edges
    const int G = 256;                 // num_graphs (reference constant)

    const float* x        = (const float*)d_in[0];
    const int*   ei       = (const int*)d_in[1];
    const int*   batch    = (const int*)d_in[2];
    const float* W_fc11   = (const float*)d_in[4];
    const float* b_fc11   = (const float*)d_in[5];
    const float* W_conv11 = (const float*)d_in[6];
    const float* b_conv11 = (const float*)d_in[7];
    const float* W_fc12   = (const float*)d_in[8];
    const float* b_fc12   = (const float*)d_in[9];
    const float* W_fc13   = (const float*)d_in[10];
    const float* b_fc13   = (const float*)d_in[11];
    const float* W_fc21   = (const float*)d_in[12];
    const float* b_fc21   = (const float*)d_in[13];
    const float* W_conv21 = (const float*)d_in[14];
    const float* b_conv21 = (const float*)d_in[15];
    const float* W_fc22   = (const float*)d_in[16];
    const float* b_fc22   = (const float*)d_in[17];
    const float* W_fc23   = (const float*)d_in[18];
    const float* b_fc23   = (const float*)d_in[19];
    const float* W_fc2    = (const float*)d_in[20];
    const float* b_fc2    = (const float*)d_in[21];

    const int* e_src = ei;
    const int* e_dst = ei + E;

    // Workspace layout (raw GEMM buffer C reused across layers; feature
    // buffer B holds x1 then x2).
    char* ws = (char*)d_ws;
    size_t off = 0;
    float* H   = (float*)(ws + off); off += align256((size_t)N * 160 * 4); // raw GEMM out
    float* X   = (float*)(ws + off); off += align256((size_t)N * 144 * 4); // x1 / x2
    float* W1c = (float*)(ws + off); off += align256((size_t)4   * 160 * 4);
    float* W2c = (float*)(ws + off); off += align256((size_t)144 * 160 * 4);
    float* ps  = (float*)(ws + off); off += align256((size_t)G * 144 * 4); // pool sums
    float* pm  = (float*)(ws + off); off += align256((size_t)G * 144 * 4); // pool maxs
    int*   pc  = (int*)  (ws + off); off += align256((size_t)G * 4);       // pool counts

    const int BT = 256;
    const int nTiles = N / 16;                          // N = 100000 -> 6250
    const int gemmBlocks = (nTiles + 7) / 8;            // 8 waves / block

    // --- pack weights for both layers ---
    pack_w_kernel<<<(4 * 160 + BT - 1) / BT, BT, 0, stream>>>(
        W_fc11, W_conv11, W_fc12, W_fc13, W1c, 4);
    pack_w_kernel<<<(144 * 160 + BT - 1) / BT, BT, 0, stream>>>(
        W_fc21, W_conv21, W_fc22, W_fc23, W2c, 144);

    // --- layer 1 ---
    gemm_wmma_kernel<<<gemmBlocks, BT, 0, stream>>>(x, 4, W1c, H, nTiles);
    build_x_kernel<<<(N + BT - 1) / BT, BT, 0, stream>>>(H, b_fc11, b_fc12, b_fc13, X, N);
    scatter_add_kernel<<<((size_t)E * 16 + BT - 1) / BT, BT, 0, stream>>>(H, e_src, e_dst, X, E);
    finish_conv_kernel<<<((size_t)N * 64 + BT - 1) / BT, BT, 0, stream>>>(X, b_conv11, N);

    // --- layer 2 (H buffer reused for raw output) ---
    gemm_wmma_kernel<<<gemmBlocks, BT, 0, stream>>>(X, 144, W2c, H, nTiles);
    build_x_kernel<<<(N + BT - 1) / BT, BT, 0, stream>>>(H, b_fc21, b_fc22, b_fc23, X, N);
    scatter_add_kernel<<<((size_t)E * 16 + BT - 1) / BT, BT, 0, stream>>>(H, e_src, e_dst, X, E);
    finish_conv_kernel<<<((size_t)N * 64 + BT - 1) / BT, BT, 0, stream>>>(X, b_conv21, N);

    // --- pooling + head ---
    pool_init_kernel<<<(G * 144 + BT - 1) / BT, BT, 0, stream>>>(ps, pm, pc);
    pool_accum_kernel<<<((size_t)N * 144 + BT - 1) / BT, BT, 0, stream>>>(X, batch, ps, pm, pc, N);
    head_kernel<<<(G + BT - 1) / BT, BT, 0, stream>>>(ps, pm, pc, W_fc2, b_fc2, (float*)d_out, G);
}